// OptimalTransportFusion_4105988735681
// MI455X (gfx1250) — compile-verified
//
#include <hip/hip_runtime.h>
#include <hip/hip_bf16.h>

typedef float v2f __attribute__((ext_vector_type(2)));
typedef float v8f __attribute__((ext_vector_type(8)));
typedef unsigned int u32x4 __attribute__((ext_vector_type(4)));
typedef int i32x8 __attribute__((ext_vector_type(8)));
typedef int i32x4 __attribute__((ext_vector_type(4)));

typedef int v4i_b __attribute__((vector_size(16)));
typedef __attribute__((address_space(1))) v4i_b gv4i;   // global int32x4
typedef __attribute__((address_space(3))) v4i_b lv4i;   // LDS    int32x4
typedef __attribute__((address_space(3))) void lds_void_t;

#define LDSS 132            // LDS row stride in floats: 132%64=4 -> conflict-free row spread
#define NEG_INV_EPS_LOG2E (-14.4269504088896341f)   // -(1/0.1)*log2(e)

#if __has_builtin(__builtin_amdgcn_global_load_async_to_lds_b128) && \
    __has_builtin(__builtin_amdgcn_s_wait_asynccnt)
#define OT_HAVE_ASYNC 1
#endif
#if __has_builtin(__builtin_amdgcn_tensor_load_to_lds) && \
    __has_builtin(__builtin_amdgcn_s_wait_tensorcnt)
#define OT_HAVE_TDM 1
#endif

// -------- staging helpers ---------------------------------------------------
__device__ __forceinline__ void stage_b128(const float* g, float* l) {
#ifdef OT_HAVE_ASYNC
  __builtin_amdgcn_global_load_async_to_lds_b128((gv4i*)(float*)g, (lv4i*)l, 0, 0);
#else
  *(float4*)l = *(const float4*)g;
#endif
}
__device__ __forceinline__ void stage_fence() {
#ifdef OT_HAVE_ASYNC
  __builtin_amdgcn_s_wait_asynccnt(0);
#endif
}

#ifdef OT_HAVE_TDM
// One TDM DMA: 128x128 f32 tile, row-major (row stride 128), into LDS with
// hardware padding of 4 DWORDs every 128 DWORDs -> LDS row stride 132 floats.
__device__ __forceinline__ void tdm_load_tile_128x128(const float* gsrc, float* lds_dst) {
  unsigned long long ga = (unsigned long long)(uintptr_t)gsrc;
  unsigned lo = (unsigned)(uintptr_t)(lds_void_t*)lds_dst;   // LDS byte offset
  u32x4 g0;
  g0.x = 1u;                                            // count=1, no gather
  g0.y = lo;                                            // lds_addr  [63:32]
  g0.z = (unsigned)(ga & 0xffffffffu);                  // global_addr[95:64]
  g0.w = (unsigned)((ga >> 32) & 0x01ffffffu) | (2u << 30);  // addr hi + type=2
  i32x8 g1;
  g1[0] = (2 << 16)      // data_size = 4B
        | (1 << 20)      // pad_enable
        | (6 << 22)      // pad_interval: 128 DWORDs
        | (3 << 25);     // pad_amount:   4 DWORDs
  g1[1] = (int)(128u << 16);   // tensor_dim0 = 128 (bits 63:48)
  g1[2] = (int)(4096u << 16);  // tensor_dim1 = 4096 (bits 111:80, low16)
  g1[3] = (int)(128u << 16);   // tile_dim0 = 128 (bits 127:112)
  g1[4] = 128;                 // tile_dim1 = 128 (bits 143:128)
  g1[5] = 128;                 // tensor_dim0_stride = 128 (bits 207:160)
  g1[6] = 0;
  g1[7] = 0;
  i32x4 z4 = {0, 0, 0, 0};
  i32x8 z8 = {0, 0, 0, 0, 0, 0, 0, 0};
  __builtin_amdgcn_tensor_load_to_lds(g0, g1, z4, z4, z8, 0);
}
#endif

// ---------------------------------------------------------------------------
// Kernel 1: projection  Y = X @ W^T + b   (rows = B*4096 = 32768, D = 128)
// blockIdx.y: 0 -> vision (Wv,bv), 1 -> clinical (Wc,bc)
// ---------------------------------------------------------------------------
__global__ __launch_bounds__(256) void ot_proj(
    const float* __restrict__ Xv, const float* __restrict__ Xc,
    const float* __restrict__ Wv, const float* __restrict__ bv,
    const float* __restrict__ Wc, const float* __restrict__ bc,
    float* __restrict__ Vout, float* __restrict__ Cout)
{
  __shared__ float Xs[128 * LDSS];
  __shared__ float Ws[128 * LDSS];

  const bool isC = (blockIdx.y != 0);
  const float* X    = isC ? Xc : Xv;
  const float* W    = isC ? Wc : Wv;
  const float* bias = isC ? bc : bv;
  float*       Y    = isC ? Cout : Vout;

  const int r0g = blockIdx.x * 128;
  const int tid = threadIdx.x;

  // stage: 128x128 X tile and full 128x128 W into LDS (async direct-to-LDS)
  for (int i = 0; i < 16; ++i) {
    int f   = tid + i * 256;      // float4 index, 4096 total
    int row = f >> 5;             // 32 float4 per row
    int cq  = (f & 31) << 2;
    stage_b128(X + (size_t)(r0g + row) * 128 + cq, &Xs[row * LDSS + cq]);
    stage_b128(W + (size_t)row * 128 + cq,         &Ws[row * LDSS + cq]);
  }
  stage_fence();
  __syncthreads();

  const int lane = tid & 31;
  const int wave = tid >> 5;
  const int hi   = lane >> 4;
  const int lo   = lane & 15;
  const int arow = wave * 16 + lo;

  v8f acc[8];
  const v8f vzero = {0.f, 0.f, 0.f, 0.f, 0.f, 0.f, 0.f, 0.f};
  for (int s = 0; s < 8; ++s) acc[s] = vzero;

  for (int t = 0; t < 32; ++t) {
    const int kb = t * 4 + hi * 2;
    v2f a = *(const v2f*)(&Xs[arow * LDSS + kb]);           // A[row][k..k+1]
    for (int s = 0; s < 8; ++s) {
      const int col = s * 16 + lo;
      v2f b = *(const v2f*)(&Ws[col * LDSS + kb]);          // B[k][col] = W[col][k]
      acc[s] = __builtin_amdgcn_wmma_f32_16x16x4_f32(
          false, a, false, b, (short)0, acc[s], false, false);
    }
  }

  for (int s = 0; s < 8; ++s) {
    const int col = s * 16 + lo;
    const float bcol = bias[col];
    for (int i = 0; i < 8; ++i) {
      const int row = r0g + wave * 16 + hi * 8 + i;
      Y[(size_t)row * 128 + col] = acc[s][i] + bcol;
    }
  }
}

// ---------------------------------------------------------------------------
// Kernel 2: row squared-norms for v and c  (one thread per row)
// ---------------------------------------------------------------------------
__global__ __launch_bounds__(256) void ot_norms(
    const float* __restrict__ V, const float* __restrict__ C,
    float* __restrict__ v2, float* __restrict__ c2)
{
  int id = blockIdx.x * 256 + threadIdx.x;     // 0..65535
  const float* src;
  float* dst;
  int r;
  if (id < 32768) { src = V; dst = v2; r = id; }
  else            { src = C; dst = c2; r = id - 32768; }
  const float4* p = (const float4*)(src + (size_t)r * 128);
  float s = 0.f;
  for (int i = 0; i < 32; ++i) {
    float4 q = p[i];
    s += q.x * q.x + q.y * q.y + q.z * q.z + q.w * q.w;
  }
  dst[r] = s;
}

// ---------------------------------------------------------------------------
// Kernel 3: fused  out = v + exp(-cdist(v,c)/eps) @ c   (flash-style over M)
// grid = (N/128, B), block = 256 (8 waves); each wave owns 16 n-rows.
// ---------------------------------------------------------------------------
__global__ __launch_bounds__(256) void ot_fuse(
    const float* __restrict__ V, const float* __restrict__ C,
    const float* __restrict__ v2g, const float* __restrict__ c2g,
    float* __restrict__ out)
{
  __shared__ float Cs[128 * LDSS];   // c tile, row-major [m][d]
  __shared__ float Ks[128 * LDSS];   // stage V tile, then per-wave K strips
  __shared__ float c2s[128];

  const int b  = blockIdx.y;
  const int n0 = blockIdx.x * 128;
  const size_t vbase  = ((size_t)b * 4096 + n0) * 128;
  const size_t cbatch = (size_t)b * 4096 * 128;

  const int tid  = threadIdx.x;
  const int lane = tid & 31;
  const int wave = tid >> 5;
  const int hi   = lane >> 4;
  const int lo   = lane & 15;

  // stage the 128x128 V tile through LDS (async), then pull A-fragments
  for (int i = 0; i < 16; ++i) {
    int f   = tid + i * 256;
    int row = f >> 5;
    int cq  = (f & 31) << 2;
    stage_b128(V + vbase + (size_t)row * 128 + cq, &Ks[row * LDSS + cq]);
  }
  stage_fence();
  __syncthreads();

  v2f vfrag[32];                                   // V A-fragments, whole kernel
  {
    const int arow = wave * 16 + lo;
    for (int t = 0; t < 32; ++t) {
      const int kb = t * 4 + hi * 2;
      vfrag[t] = *(const v2f*)(&Ks[arow * LDSS + kb]);
    }
  }
  float vv2[8];                                    // ||v||^2 for my C/D rows
  for (int i = 0; i < 8; ++i)
    vv2[i] = v2g[(size_t)b * 4096 + n0 + wave * 16 + hi * 8 + i];

  v8f acc[8];
  const v8f vzero = {0.f, 0.f, 0.f, 0.f, 0.f, 0.f, 0.f, 0.f};
  for (int s = 0; s < 8; ++s) acc[s] = vzero;

  float* Kw = &Ks[wave * 16 * LDSS];               // this wave's 16x128 K strip

  for (int mt = 0; mt < 32; ++mt) {
    __syncthreads();                               // Cs consumers of prev iter done
    const int m0 = mt * 128;
#ifdef OT_HAVE_TDM
    if (tid < 32) {                                // wave 0 issues the tile DMA
      tdm_load_tile_128x128(C + cbatch + (size_t)m0 * 128, Cs);
      __builtin_amdgcn_s_wait_tensorcnt(0);
    }
#else
    for (int i = 0; i < 16; ++i) {
      int f   = tid + i * 256;
      int row = f >> 5;
      int cq  = (f & 31) << 2;
      stage_b128(C + cbatch + (size_t)(m0 + row) * 128 + cq, &Cs[row * LDSS + cq]);
    }
    stage_fence();
#endif
    if (tid < 128) c2s[tid] = c2g[(size_t)b * 4096 + m0 + tid];
    __syncthreads();

    // --- S = V . Cs^T, then K = exp(-sqrt(max(v2+c2-2S,0))/eps) -> LDS strip
    for (int ms = 0; ms < 8; ++ms) {
      v8f s = vzero;
      const int bcol = ms * 16 + lo;
      for (int t = 0; t < 32; ++t) {
        const int kb = t * 4 + hi * 2;
        v2f bf = *(const v2f*)(&Cs[bcol * LDSS + kb]);    // B[k][m]=C[m][k]
        s = __builtin_amdgcn_wmma_f32_16x16x4_f32(
            false, vfrag[t], false, bf, (short)0, s, false, false);
      }
      const float c2v = c2s[bcol];
      for (int i = 0; i < 8; ++i) {
        float d2   = vv2[i] + c2v - 2.0f * s[i];
        float cost = __builtin_amdgcn_sqrtf(fmaxf(d2, 0.0f));
        float kv   = __builtin_amdgcn_exp2f(cost * NEG_INV_EPS_LOG2E);
        Kw[(hi * 8 + i) * LDSS + bcol] = kv;
      }
    }
    __syncthreads();                               // K strips committed

    // --- acc += K . Cs   (A = K strip rows, B = Cs columns)
    for (int t = 0; t < 32; ++t) {
      const int kb = t * 4 + hi * 2;
      v2f a = *(const v2f*)(&Kw[lo * LDSS + kb]);          // A[n][m..m+1]
      for (int dt = 0; dt < 8; ++dt) {
        const int col = dt * 16 + lo;
        v2f bf;
        bf.x = Cs[(kb + 0) * LDSS + col];                  // B[m][d]
        bf.y = Cs[(kb + 1) * LDSS + col];
        acc[dt] = __builtin_amdgcn_wmma_f32_16x16x4_f32(
            false, a, false, bf, (short)0, acc[dt], false, false);
      }
    }
  }

  // epilogue: out = v + acc
  for (int dt = 0; dt < 8; ++dt) {
    const int col = dt * 16 + lo;
    for (int i = 0; i < 8; ++i) {
      const size_t idx = vbase + (size_t)(wave * 16 + hi * 8 + i) * 128 + col;
      out[idx] = V[idx] + acc[dt][i];
    }
  }
}

// ---------------------------------------------------------------------------
extern "C" void kernel_launch(void* const* d_in, const int* in_sizes, int n_in,
                              void* d_out, int out_size, void* d_ws, size_t ws_size,
                              hipStream_t stream) {
  const float* Xv = (const float*)d_in[0];   // [8,4096,128]
  const float* Xc = (const float*)d_in[1];   // [8,4096,128]
  const float* Wv = (const float*)d_in[2];   // [128,128]
  const float* bv = (const float*)d_in[3];   // [128]
  const float* Wc = (const float*)d_in[4];   // [128,128]
  const float* bc = (const float*)d_in[5];   // [128]
  float* out = (float*)d_out;

  float* ws = (float*)d_ws;
  float* Vp = ws;                     // 8*4096*128 = 4194304 floats
  float* Cp = Vp + 4194304;           // 4194304 floats
  float* v2 = Cp + 4194304;           // 32768 floats
  float* c2 = v2 + 32768;             // 32768 floats

  ot_proj<<<dim3(256, 2), dim3(256), 0, stream>>>(Xv, Xc, Wv, bv, Wc, bc, Vp, Cp);
  ot_norms<<<dim3(256), dim3(256), 0, stream>>>(Vp, Cp, v2, c2);
  ot_fuse<<<dim3(32, 8), dim3(256), 0, stream>>>(Vp, Cp, v2, c2, out);
}